// MultiHeadAttention_47175920780067
// MI455X (gfx1250) — compile-verified
//
#include <hip/hip_runtime.h>

typedef float v2f __attribute__((ext_vector_type(2)));
typedef float v8f __attribute__((ext_vector_type(8)));
typedef int v4i __attribute__((__vector_size__(16)));

#define NB 8
#define NC 256
#define NL 25600
#define TL 64
#define TLP 68 /* row pitch: 16B-aligned (68%4==0) and conflict-free fragment reads */
#define NLT (NL / TL) /* 400 */

#if defined(__has_builtin)
#if __has_builtin(__builtin_amdgcn_global_load_async_to_lds_b128)
#define HAVE_ASYNC 1
#endif
#endif

typedef __attribute__((address_space(1))) v4i gv4i_t;
typedef __attribute__((address_space(3))) v4i lv4i_t;

// 16-byte global -> LDS copy; async DMA (ASYNCcnt) when the toolchain has it.
__device__ __forceinline__ void cp16(const float* __restrict__ g,
                                     float* __restrict__ l) {
#ifdef HAVE_ASYNC
  float* gg = const_cast<float*>(g);
  __builtin_amdgcn_global_load_async_to_lds_b128((gv4i_t*)gg, (lv4i_t*)l, 0, 0);
#else
  const float4 v = *(const float4*)g;
  l[0] = v.x; l[1] = v.y; l[2] = v.z; l[3] = v.w;
#endif
}

__device__ __forceinline__ void wait_async() {
#ifdef HAVE_ASYNC
#if __has_builtin(__builtin_amdgcn_s_wait_asynccnt)
  __builtin_amdgcn_s_wait_asynccnt(0);
#else
  asm volatile("s_wait_asynccnt 0x0" ::: "memory");
#endif
#endif
}

// One [NC x TL] tile (row pitch TLP) from global to LDS.
__device__ __forceinline__ void tile_load(const float* __restrict__ src,
                                          float* __restrict__ dst, size_t base,
                                          int l0, int tid, int nthr) {
  for (int i = tid; i < NC * (TL / 4); i += nthr) {
    const int c = i >> 4;
    const int x = (i & 15) * 4;
    cp16(src + base + (size_t)c * NL + l0 + x, dst + c * TLP + x);
  }
}

// D = A(16x4, f32) * B(4x16, f32) + C(16x16, f32)
__device__ __forceinline__ v8f wmma4(v2f a, v2f b, v8f c) {
  return __builtin_amdgcn_wmma_f32_16x16x4_f32(false, a, false, b, (short)0, c,
                                               false, false);
}

__global__ void k_zero(float* __restrict__ p, int n) {
  const int i = blockIdx.x * blockDim.x + threadIdx.x;
  if (i < n) p[i] = 0.f;
}

// G[b] += X_q X_k^T over a slice of L; also per-channel row-sums of X_q, X_k.
__global__ __launch_bounds__(1024) void k_gram(
    const float* __restrict__ Q, const float* __restrict__ K,
    float* __restrict__ G, float* __restrict__ sq, float* __restrict__ sk) {
  __shared__ float sQ[2][NC * TLP];
  __shared__ float sK[2][NC * TLP];
  const int b = blockIdx.y;
  const int tid = threadIdx.x;
  const int lane = tid & 31;
  const int wave = tid >> 5;
  const int l16 = lane & 15;
  const int khi = (lane & 16) ? 2 : 0;
  const int mhi = (lane & 16) ? 8 : 0;
  const int m0 = (wave >> 1) * 16;   // 16 M-slabs
  const int nb = (wave & 1) * 8;     // half of the 16 N-tiles

  v8f acc[8];
#pragma unroll
  for (int t = 0; t < 8; ++t) {
#pragma unroll
    for (int r = 0; r < 8; ++r) acc[t][r] = 0.f;
  }

  float rq = 0.f, rk = 0.f;
  const int cS = tid & 255;
  const int pS = (tid >> 8) * 16;

  const size_t base = (size_t)b * NC * NL;

  int lt = blockIdx.x;
  int buf = 0;
  tile_load(Q, sQ[0], base, lt * TL, tid, 1024);
  tile_load(K, sK[0], base, lt * TL, tid, 1024);

  for (; lt < NLT; lt += gridDim.x, buf ^= 1) {
    wait_async();
    __syncthreads();
    const int nlt = lt + gridDim.x;
    if (nlt < NLT) {
      tile_load(Q, sQ[buf ^ 1], base, nlt * TL, tid, 1024);
      tile_load(K, sK[buf ^ 1], base, nlt * TL, tid, 1024);
    }
    const float* __restrict__ bQ = sQ[buf];
    const float* __restrict__ bK = sK[buf];

    {
      float aq = 0.f, ak = 0.f;
#pragma unroll
      for (int x = 0; x < 16; ++x) {
        aq += bQ[cS * TLP + pS + x];
        ak += bK[cS * TLP + pS + x];
      }
      rq += aq;
      rk += ak;
    }

    for (int kk = 0; kk < TL; kk += 4) {
      v2f A;
      A.x = bQ[(m0 + l16) * TLP + kk + khi];
      A.y = bQ[(m0 + l16) * TLP + kk + khi + 1];
      v2f Bv[8];
#pragma unroll
      for (int ni = 0; ni < 8; ++ni) {
        const int n0 = (nb + ni) * 16;
        Bv[ni].x = bK[(n0 + l16) * TLP + kk + khi];
        Bv[ni].y = bK[(n0 + l16) * TLP + kk + khi + 1];
      }
#pragma unroll
      for (int ni = 0; ni < 8; ++ni) acc[ni] = wmma4(A, Bv[ni], acc[ni]);
    }
    __syncthreads();
  }

  float* Gb = G + (size_t)b * NC * NC;
#pragma unroll
  for (int ni = 0; ni < 8; ++ni) {
    const int n0 = (nb + ni) * 16;
#pragma unroll
    for (int r = 0; r < 8; ++r)
      atomicAdd(&Gb[(m0 + r + mhi) * NC + n0 + l16], acc[ni][r]);
  }
  atomicAdd(&sq[b * NC + cS], rq);
  atomicAdd(&sk[b * NC + cS], rk);
}

// scores[b,h] = (Wq_h G_b Wk_h^T + bq*(Wk sk) + bk*(Wq sq) + L*bq*bk) / 16
__global__ __launch_bounds__(256) void k_scores(
    const float* __restrict__ G, const float* __restrict__ sq,
    const float* __restrict__ sk, const float* __restrict__ Wq,
    const float* __restrict__ bq, const float* __restrict__ Wk,
    const float* __restrict__ bk, float* __restrict__ scores) {
  __shared__ float tmp[32 * 256];
  const int h = blockIdx.x, b = blockIdx.y;
  const float* Gb = G + (size_t)b * NC * NC;
  for (int idx = threadIdx.x; idx < 32 * 256; idx += 256) {
    const int d = idx >> 8, c2 = idx & 255;
    const float* wr = Wq + (h * 32 + d) * NC;
    float s = 0.f;
    for (int c = 0; c < NC; ++c) s += wr[c] * Gb[c * NC + c2];
    tmp[idx] = s;
  }
  __syncthreads();
  for (int idx = threadIdx.x; idx < 1024; idx += 256) {
    const int d = idx >> 5, e = idx & 31;
    const int D = h * 32 + d, E = h * 32 + e;
    float s = 0.f, wqs = 0.f, wks = 0.f;
    for (int c = 0; c < NC; ++c) {
      s += tmp[d * 256 + c] * Wk[E * NC + c];
      wqs += Wq[D * NC + c] * sq[b * NC + c];
      wks += Wk[E * NC + c] * sk[b * NC + c];
    }
    s += bq[D] * wks + bk[E] * wqs + (float)NL * bq[D] * bk[E];
    scores[((size_t)(b * 8 + h) * 32 + d) * 32 + e] = s * (1.0f / 16.0f);
  }
}

// in-place softmax over last dim (32); one wave32 per row
__global__ void k_softmax(float* __restrict__ scores) {
  const int bh = blockIdx.x;
  const int e = threadIdx.x, d = threadIdx.y;
  float* row = scores + (size_t)bh * 1024 + d * 32;
  const float v = row[e];
  float m = v;
#pragma unroll
  for (int o = 16; o > 0; o >>= 1) m = fmaxf(m, __shfl_xor(m, o, 32));
  const float ex = __expf(v - m);
  float s = ex;
#pragma unroll
  for (int o = 16; o > 0; o >>= 1) s += __shfl_xor(s, o, 32);
  row[e] = ex / s;
}

// Weff[b] = blockdiag(attn_b) @ Wv ; beff[b] = blockdiag(attn_b) @ bv
__global__ __launch_bounds__(256) void k_weff(
    const float* __restrict__ attn, const float* __restrict__ Wv,
    const float* __restrict__ bv, float* __restrict__ Weff,
    float* __restrict__ beff) {
  const int b = blockIdx.x;
  for (int idx = threadIdx.x; idx < NC * NC; idx += 256) {
    const int D = idx >> 8, c = idx & 255;
    const int h = D >> 5, d = D & 31;
    const float* ar = attn + ((size_t)(b * 8 + h) * 32 + d) * 32;
    float s = 0.f;
    for (int e = 0; e < 32; ++e) s += ar[e] * Wv[(h * 32 + e) * NC + c];
    Weff[((size_t)b * NC + D) * NC + c] = s;
  }
  for (int D = threadIdx.x; D < NC; D += 256) {
    const int h = D >> 5, d = D & 31;
    const float* ar = attn + ((size_t)(b * 8 + h) * 32 + d) * 32;
    float s = 0.f;
    for (int e = 0; e < 32; ++e) s += ar[e] * bv[h * 32 + e];
    beff[b * NC + D] = s;
  }
}

// out[b] = Weff[b] @ key[b] + beff[b]; fused per-channel sum / sumsq
__global__ __launch_bounds__(512) void k_attn_out(
    const float* __restrict__ K, const float* __restrict__ Weff,
    const float* __restrict__ beff, float* __restrict__ Ob,
    float* __restrict__ sum, float* __restrict__ sumsq) {
  __shared__ float sX[2][NC * TLP];
  const int b = blockIdx.y;
  const int tid = threadIdx.x;
  const int lane = tid & 31;
  const int wave = tid >> 5;
  const int l16 = lane & 15;
  const int khi = (lane & 16) ? 2 : 0;
  const int mhi = (lane & 16) ? 8 : 0;
  const int m0 = wave * 16;

  const float* Wb = Weff + (size_t)b * NC * NC;
  float cb[8];
#pragma unroll
  for (int r = 0; r < 8; ++r) cb[r] = beff[b * NC + m0 + r + mhi];

  float aS[4][8], aQ[4][8];
#pragma unroll
  for (int ni = 0; ni < 4; ++ni) {
#pragma unroll
    for (int r = 0; r < 8; ++r) { aS[ni][r] = 0.f; aQ[ni][r] = 0.f; }
  }

  const size_t base = (size_t)b * NC * NL;

  int lt = blockIdx.x;
  int buf = 0;
  tile_load(K, sX[0], base, lt * TL, tid, 512);

  for (; lt < NLT; lt += gridDim.x, buf ^= 1) {
    wait_async();
    __syncthreads();
    const int nlt = lt + gridDim.x;
    if (nlt < NLT) tile_load(K, sX[buf ^ 1], base, nlt * TL, tid, 512);
    const float* __restrict__ sb = sX[buf];
    const int l0 = lt * TL;

    v8f acc[4];
#pragma unroll
    for (int ni = 0; ni < 4; ++ni) {
#pragma unroll
      for (int r = 0; r < 8; ++r) acc[ni][r] = cb[r];
    }

    for (int kt = 0; kt < NC; kt += 32) {
      v2f Af[8];
#pragma unroll
      for (int j = 0; j < 8; ++j) {
        const int kk = kt + j * 4;
        Af[j].x = Wb[(m0 + l16) * NC + kk + khi];
        Af[j].y = Wb[(m0 + l16) * NC + kk + khi + 1];
      }
#pragma unroll
      for (int j = 0; j < 8; ++j) {
        const int kk = kt + j * 4;
        v2f Bv[4];
#pragma unroll
        for (int ni = 0; ni < 4; ++ni) {
          Bv[ni].x = sb[(kk + khi) * TLP + ni * 16 + l16];
          Bv[ni].y = sb[(kk + khi + 1) * TLP + ni * 16 + l16];
        }
#pragma unroll
        for (int ni = 0; ni < 4; ++ni) acc[ni] = wmma4(Af[j], Bv[ni], acc[ni]);
      }
    }

#pragma unroll
    for (int ni = 0; ni < 4; ++ni) {
#pragma unroll
      for (int r = 0; r < 8; ++r) {
        const float v = acc[ni][r];
        Ob[base + (size_t)(m0 + r + mhi) * NL + l0 + ni * 16 + l16] = v;
        aS[ni][r] += v;
        aQ[ni][r] += v * v;
      }
    }
  }

#pragma unroll
  for (int ni = 0; ni < 4; ++ni) {
#pragma unroll
    for (int r = 0; r < 8; ++r) {
      float s = aS[ni][r], q = aQ[ni][r];
#pragma unroll
      for (int o = 1; o < 16; o <<= 1) {
        s += __shfl_xor(s, o, 32);
        q += __shfl_xor(q, o, 32);
      }
      if (l16 == 0) {
        atomicAdd(&sum[b * NC + m0 + r + mhi], s);
        atomicAdd(&sumsq[b * NC + m0 + r + mhi], q);
      }
    }
  }
}

__global__ void k_stats(const float* __restrict__ sum,
                        const float* __restrict__ sumsq, float* __restrict__ mu,
                        float* __restrict__ rstd) {
  const int i = blockIdx.x * blockDim.x + threadIdx.x;
  if (i < NB * NC) {
    const float m = sum[i] * (1.f / (float)NL);
    float v = sumsq[i] * (1.f / (float)NL) - m * m;
    v = fmaxf(v, 0.f);
    mu[i] = m;
    rstd[i] = rsqrtf(v + 1e-5f);
  }
}

// y[b] = Wo @ ((out - mu)*rstd*gamma[l] + beta[l]) + bo
__global__ __launch_bounds__(512) void k_final(
    const float* __restrict__ Ob, const float* __restrict__ mu,
    const float* __restrict__ rstd, const float* __restrict__ gamma,
    const float* __restrict__ beta, const float* __restrict__ Wo,
    const float* __restrict__ bo, float* __restrict__ Y) {
  __shared__ float sX[2][NC * TLP];
  const int b = blockIdx.y;
  const int tid = threadIdx.x;
  const int lane = tid & 31;
  const int wave = tid >> 5;
  const int l16 = lane & 15;
  const int khi = (lane & 16) ? 2 : 0;
  const int mhi = (lane & 16) ? 8 : 0;
  const int m0 = wave * 16;

  float cb[8];
#pragma unroll
  for (int r = 0; r < 8; ++r) cb[r] = bo[m0 + r + mhi];

  const size_t base = (size_t)b * NC * NL;

  int lt = blockIdx.x;
  int buf = 0;
  tile_load(Ob, sX[0], base, lt * TL, tid, 512);

  for (; lt < NLT; lt += gridDim.x, buf ^= 1) {
    wait_async();
    __syncthreads();
    const int nlt = lt + gridDim.x;
    if (nlt < NLT) tile_load(Ob, sX[buf ^ 1], base, nlt * TL, tid, 512);
    float* __restrict__ sb = sX[buf];
    const int l0 = lt * TL;

    // normalize the raw tile in LDS: (x - mu_c)*rstd_c*gamma[l] + beta[l]
    for (int i = tid; i < NC * (TL / 4); i += 512) {
      const int c = i >> 4;
      const int x = (i & 15) * 4;
      const float m = mu[b * NC + c];
      const float rs = rstd[b * NC + c];
      const float* g = gamma + l0 + x;
      const float* bt = beta + l0 + x;
      float* dp = &sb[c * TLP + x];
#pragma unroll
      for (int j = 0; j < 4; ++j) dp[j] = (dp[j] - m) * rs * g[j] + bt[j];
    }
    __syncthreads();

    v8f acc[4];
#pragma unroll
    for (int ni = 0; ni < 4; ++ni) {
#pragma unroll
      for (int r = 0; r < 8; ++r) acc[ni][r] = cb[r];
    }

    for (int kt = 0; kt < NC; kt += 32) {
      v2f Af[8];
#pragma unroll
      for (int j = 0; j < 8; ++j) {
        const int kk = kt + j * 4;
        Af[j].x = Wo[(m0 + l16) * NC + kk + khi];
        Af[j].y = Wo[(m0 + l16) * NC + kk + khi + 1];
      }
#pragma unroll
      for (int j = 0; j < 8; ++j) {
        const int kk = kt + j * 4;
        v2f Bv[4];
#pragma unroll
        for (int ni = 0; ni < 4; ++ni) {
          Bv[ni].x = sb[(kk + khi) * TLP + ni * 16 + l16];
          Bv[ni].y = sb[(kk + khi + 1) * TLP + ni * 16 + l16];
        }
#pragma unroll
        for (int ni = 0; ni < 4; ++ni) acc[ni] = wmma4(Af[j], Bv[ni], acc[ni]);
      }
    }

#pragma unroll
    for (int ni = 0; ni < 4; ++ni) {
#pragma unroll
      for (int r = 0; r < 8; ++r)
        Y[base + (size_t)(m0 + r + mhi) * NL + l0 + ni * 16 + l16] =
            acc[ni][r];
    }
  }
}

extern "C" void kernel_launch(void* const* d_in, const int* in_sizes, int n_in,
                              void* d_out, int out_size, void* d_ws,
                              size_t ws_size, hipStream_t stream) {
  (void)in_sizes; (void)n_in; (void)out_size; (void)ws_size;
  const float* query = (const float*)d_in[0];
  const float* key   = (const float*)d_in[1];
  const float* Wq = (const float*)d_in[2];
  const float* bq = (const float*)d_in[3];
  const float* Wk = (const float*)d_in[4];
  const float* bk = (const float*)d_in[5];
  const float* Wv = (const float*)d_in[6];
  const float* bv = (const float*)d_in[7];
  const float* Wo = (const float*)d_in[8];
  const float* bo = (const float*)d_in[9];
  const float* gamma = (const float*)d_in[10];
  const float* beta  = (const float*)d_in[11];

  float* ws = (float*)d_ws;
  float* G      = ws;                   // 8*256*256 = 524288
  float* sq     = G + 524288;           // 2048
  float* sk     = sq + 2048;            // 2048
  float* sum    = sk + 2048;            // 2048
  float* sumsq  = sum + 2048;           // 2048
  float* mu     = sumsq + 2048;         // 2048
  float* rstd   = mu + 2048;            // 2048
  float* scores = rstd + 2048;          // 8*8*32*32 = 65536
  float* Weff   = scores + 65536;       // 524288
  float* beff   = Weff + 524288;        // 2048
  float* Ob     = beff + 2048;          // 8*256*25600

  float* Y = (float*)d_out;

  const int nzero = 524288 + 4 * 2048;  // G, sq, sk, sum, sumsq
  k_zero<<<dim3((nzero + 255) / 256), 256, 0, stream>>>(ws, nzero);
  k_gram<<<dim3(32, NB), 1024, 0, stream>>>(query, key, G, sq, sk);
  k_scores<<<dim3(8, NB), 256, 0, stream>>>(G, sq, sk, Wq, bq, Wk, bk, scores);
  k_softmax<<<dim3(NB * 8), dim3(32, 32), 0, stream>>>(scores);
  k_weff<<<dim3(NB), 256, 0, stream>>>(scores, Wv, bv, Weff, beff);
  k_attn_out<<<dim3(50, NB), 512, 0, stream>>>(key, Weff, beff, Ob, sum, sumsq);
  k_stats<<<dim3(8), 256, 0, stream>>>(sum, sumsq, mu, rstd);
  k_final<<<dim3(50, NB), 512, 0, stream>>>(Ob, mu, rstd, gamma, beta, Wo, bo, Y);
}